// GINLayer_43061342110476
// MI455X (gfx1250) — compile-verified
//
#include <hip/hip_runtime.h>

// GIN layer: out = x + relu(BN(relu((x + segsum(x[src]->dst)) @ w1 + b1) @ w2 + b2))
// N=50000 nodes, E=600000 edges, D=128. Memory-bound (scatter ~614MB of traffic);
// GEMMs done with fp32 WMMA (v_wmma_f32_16x16x4_f32) to match reference precision.

typedef __attribute__((ext_vector_type(2))) float v2f;
typedef __attribute__((ext_vector_type(8))) float v8f;

#define N_NODES 50000
#define N_EDGES 600000
#define DIM     128
#define BN_EPS  1e-5f
#define AB_STRIDE 132   // LDS row stride (floats): 132 -> banks step by 4, no conflicts

// ---------------------------------------------------------------- zero ws
__global__ __launch_bounds__(256) void gin_zero(float* __restrict__ agg,
                                                float* __restrict__ stats) {
    size_t i = (size_t)blockIdx.x * 256 + threadIdx.x;      // one float4 each
    if (i < (size_t)N_NODES * DIM / 4) {
        ((float4*)agg)[i] = make_float4(0.f, 0.f, 0.f, 0.f);
    }
    if (blockIdx.x == 0 && threadIdx.x < 256) stats[threadIdx.x] = 0.f; // sum+sumsq
}

// ---------------------------------------------------------------- scatter-add
__global__ __launch_bounds__(256) void gin_scatter(const float* __restrict__ x,
                                                   const int* __restrict__ src,
                                                   const int* __restrict__ dst,
                                                   float* __restrict__ agg) {
    size_t gid = (size_t)blockIdx.x * 256 + threadIdx.x;    // E*32 threads, 4 floats each
    if (gid >= (size_t)N_EDGES * 32) return;
    int e  = (int)(gid >> 5);
    int c4 = (int)(gid & 31) << 2;
    int s = src[e];
    int d = dst[e];
    float4 v = *(const float4*)(x + (size_t)s * DIM + c4);
    float* p = agg + (size_t)d * DIM + c4;
    __hip_atomic_fetch_add(p + 0, v.x, __ATOMIC_RELAXED, __HIP_MEMORY_SCOPE_AGENT);
    __hip_atomic_fetch_add(p + 1, v.y, __ATOMIC_RELAXED, __HIP_MEMORY_SCOPE_AGENT);
    __hip_atomic_fetch_add(p + 2, v.z, __ATOMIC_RELAXED, __HIP_MEMORY_SCOPE_AGENT);
    __hip_atomic_fetch_add(p + 3, v.w, __ATOMIC_RELAXED, __HIP_MEMORY_SCOPE_AGENT);
}

// ---------------------------------------------------------------- WMMA GEMM band
// One block = one 16-row band of the [N,128] activation; 8 waves cover the 8
// 16-col tiles. A comes from LDS (x+agg or h1), B streamed from L2-hot weights.
template <bool ADD_AGG, bool FUSE_RELU, bool DO_STATS>
__global__ __launch_bounds__(256) void gin_gemm(const float* __restrict__ in0,
                                                const float* __restrict__ in1,
                                                const float* __restrict__ w,
                                                const float* __restrict__ bias,
                                                float* __restrict__ out,
                                                float* __restrict__ stat_sum,
                                                float* __restrict__ stat_sq) {
    __shared__ float aband[16 * AB_STRIDE];
    __shared__ float red_s[256];
    __shared__ float red_q[256];

    const int t    = threadIdx.x;
    const int row0 = blockIdx.x * 16;

    // cooperative load of the 16x128 A band into LDS (forming x + agg for GEMM1)
    for (int i = t; i < 16 * DIM; i += 256) {
        int r = i >> 7;
        int k = i & 127;
        float v = in0[(size_t)(row0 + r) * DIM + k];
        if (ADD_AGG) v += in1[(size_t)(row0 + r) * DIM + k];
        aband[r * AB_STRIDE + k] = v;
    }
    __syncthreads();

    const int wave  = t >> 5;
    const int lane  = t & 31;
    const int col   = lane & 15;     // N within tile / M row for A fetch
    const int hi    = lane >> 4;     // K-half select (ISA 16x4 f32 A/B layout)
    const int n0    = wave * 16;
    const int khalf = hi * 2;

    v8f acc = {};
    #pragma unroll 4
    for (int k0 = 0; k0 < DIM; k0 += 4) {
        v2f a = *(const v2f*)&aband[col * AB_STRIDE + k0 + khalf];
        v2f b;
        b.x = w[(size_t)(k0 + khalf) * DIM + n0 + col];
        b.y = w[(size_t)(k0 + khalf + 1) * DIM + n0 + col];
        acc = __builtin_amdgcn_wmma_f32_16x16x4_f32(
            /*neg_a=*/false, a, /*neg_b=*/false, b,
            /*c_mod=*/(short)0, acc, /*reuse_a=*/false, /*reuse_b=*/false);
    }

    const float bv = bias[n0 + col];
    float lsum = 0.f, lsq = 0.f;
    #pragma unroll
    for (int r = 0; r < 8; ++r) {
        float v = acc[r] + bv;
        if (FUSE_RELU) v = fmaxf(v, 0.f);
        const int m = r + hi * 8;    // lanes 0-15: rows 0-7, lanes 16-31: rows 8-15
        out[(size_t)(row0 + m) * DIM + n0 + col] = v;
        if (DO_STATS) { lsum += v; lsq += v * v; }
    }

    if (DO_STATS) {
        red_s[t] = lsum;
        red_q[t] = lsq;
        __syncthreads();
        if (hi == 0) {               // lanes 0-15: combine with partner lane+16
            float s = red_s[t] + red_s[t + 16];
            float q = red_q[t] + red_q[t + 16];
            int c = n0 + col;
            __hip_atomic_fetch_add(&stat_sum[c], s, __ATOMIC_RELAXED, __HIP_MEMORY_SCOPE_AGENT);
            __hip_atomic_fetch_add(&stat_sq[c],  q, __ATOMIC_RELAXED, __HIP_MEMORY_SCOPE_AGENT);
        }
    }
}

// ---------------------------------------------------------------- BN finalize
__global__ __launch_bounds__(128) void gin_bnstats(const float* __restrict__ ssum,
                                                   const float* __restrict__ ssq,
                                                   const float* __restrict__ gamma,
                                                   const float* __restrict__ beta,
                                                   float* __restrict__ scale,
                                                   float* __restrict__ shift) {
    int c = threadIdx.x;
    float inv_n = 1.0f / (float)N_NODES;
    float mean  = ssum[c] * inv_n;
    float var   = ssq[c] * inv_n - mean * mean;     // biased variance
    float sc    = rsqrtf(var + BN_EPS) * gamma[c];
    scale[c] = sc;
    shift[c] = beta[c] - mean * sc;
}

// ---------------------------------------------------------------- BN + residual
__global__ __launch_bounds__(256) void gin_bnres(const float* __restrict__ x,
                                                 float* __restrict__ h,
                                                 const float* __restrict__ scale,
                                                 const float* __restrict__ shift) {
    size_t i = (size_t)blockIdx.x * 256 + threadIdx.x;   // one float4
    if (i >= (size_t)N_NODES * DIM / 4) return;
    int c4 = (int)(i & (DIM / 4 - 1)) << 2;
    float4 hv = ((const float4*)h)[i];
    float4 xv = ((const float4*)x)[i];
    float4 o;
    o.x = xv.x + fmaxf(hv.x * scale[c4 + 0] + shift[c4 + 0], 0.f);
    o.y = xv.y + fmaxf(hv.y * scale[c4 + 1] + shift[c4 + 1], 0.f);
    o.z = xv.z + fmaxf(hv.z * scale[c4 + 2] + shift[c4 + 2], 0.f);
    o.w = xv.w + fmaxf(hv.w * scale[c4 + 3] + shift[c4 + 3], 0.f);
    ((float4*)h)[i] = o;
}

// ---------------------------------------------------------------- launch
extern "C" void kernel_launch(void* const* d_in, const int* in_sizes, int n_in,
                              void* d_out, int out_size, void* d_ws, size_t ws_size,
                              hipStream_t stream) {
    const float* x     = (const float*)d_in[0];
    const int*   ei    = (const int*)d_in[1];     // [2,E]: src then dst
    const float* w1    = (const float*)d_in[2];
    const float* b1    = (const float*)d_in[3];
    const float* w2    = (const float*)d_in[4];
    const float* b2    = (const float*)d_in[5];
    const float* gamma = (const float*)d_in[6];
    const float* beta  = (const float*)d_in[7];
    float* out = (float*)d_out;

    // ws layout (floats): [agg|h1 alias : N*D][sum:128][sumsq:128][scale:128][shift:128]
    float* agg   = (float*)d_ws;
    float* stats = agg + (size_t)N_NODES * DIM;
    float* ssum  = stats;
    float* ssq   = stats + 128;
    float* scale = stats + 256;
    float* shift = stats + 384;

    const int vec4_blocks = (int)(((size_t)N_NODES * DIM / 4 + 255) / 256);   // 6250
    gin_zero<<<vec4_blocks, 256, 0, stream>>>(agg, stats);

    const int scat_blocks = (int)(((size_t)N_EDGES * 32 + 255) / 256);        // 75000
    gin_scatter<<<scat_blocks, 256, 0, stream>>>(x, ei, ei + N_EDGES, agg);

    const int band_blocks = N_NODES / 16;                                     // 3125
    // h1 = relu((x+agg)@w1 + b1), written back over agg (per-block safe alias)
    gin_gemm<true, true, false><<<band_blocks, 256, 0, stream>>>(
        x, agg, w1, b1, agg, nullptr, nullptr);
    // h2 = h1@w2 + b2 -> d_out, fused per-channel sum/sumsq
    gin_gemm<false, false, true><<<band_blocks, 256, 0, stream>>>(
        agg, nullptr, w2, b2, out, ssum, ssq);

    gin_bnstats<<<1, 128, 0, stream>>>(ssum, ssq, gamma, beta, scale, shift);
    gin_bnres<<<vec4_blocks, 256, 0, stream>>>(x, out, scale, shift);
}